// BERT_pool_mutil_avr_18631568130146
// MI455X (gfx1250) — compile-verified
//
#include <hip/hip_runtime.h>

typedef __attribute__((ext_vector_type(2))) float v2f;
typedef __attribute__((ext_vector_type(8))) float v8f;
typedef __attribute__((ext_vector_type(4))) int   v4i;

#define WMMA_F32(a, b, c) \
  __builtin_amdgcn_wmma_f32_16x16x4_f32(false, (a), false, (b), (short)0, (c), false, false)

// CDNA5 async global->LDS copy (ASYNCcnt-tracked), if the toolchain has it.
#if defined(__has_builtin)
#if __has_builtin(__builtin_amdgcn_global_load_async_to_lds_b128)
#define USE_ASYNC_COPY 1
#endif
#endif
#ifndef USE_ASYNC_COPY
#define USE_ASYNC_COPY 0
#endif

#if defined(__has_builtin)
#if __has_builtin(__builtin_amdgcn_s_wait_asynccnt)
#define WAIT_ASYNC0() __builtin_amdgcn_s_wait_asynccnt(0)
#else
#define WAIT_ASYNC0() asm volatile("s_wait_asynccnt 0x0" ::: "memory")
#endif
#else
#define WAIT_ASYNC0() asm volatile("s_wait_asynccnt 0x0" ::: "memory")
#endif

// Exact pointer types the async builtin expects (param 1 prints as
// "int __vector(4) __device__ *" == v4i addrspace(1)*).
#define ASYNC_GPTR(p) ((__attribute__((address_space(1))) v4i*)(p))
#define ASYNC_SPTR(p) ((__attribute__((address_space(3))) v4i*)(p))

// Problem dims (fixed by reference setup_inputs)
constexpr int Bc   = 4;
constexpr int Dc   = 1024;
constexpr int Tc   = 2048;
constexpr int NROI = 128;
constexpr int Hc   = 8;
constexpr int CG   = 256;   // D / hec

// LDS A-tile row stride (floats): 72*4B is 16B-aligned for b128 stores, and
// 72 mod 64 = 8 skews banks so lo/hi half-wave fragment reads are disjoint.
constexpr int LSTR = 72;

// ---------------------------------------------------------------------------
// K-pair interleaved transpose: src W [R rows (out), C cols (k)] row-major ->
// dst WtP[((k/2)*R + r)*2 + (k&1)] so a WMMA B-fragment's (k, k+1) pair is one
// contiguous 8-byte load per lane, coalesced across the 16 lanes of a tile.
// ---------------------------------------------------------------------------
__global__ void transpose_pair_kernel(const float* __restrict__ src,
                                      float* __restrict__ dst, int R, int C) {
  size_t idx = (size_t)blockIdx.x * blockDim.x + threadIdx.x;
  size_t total = (size_t)R * C;
  if (idx >= total) return;
  int r = (int)(idx / C);
  int c = (int)(idx % C);
  dst[((size_t)(c >> 1) * R + r) * 2 + (c & 1)] = src[idx];
}

// ---------------------------------------------------------------------------
// q = cls @ Wq^T + bq ; nq[b,h] = ||q[b, h*128 : (h+1)*128]||
// ---------------------------------------------------------------------------
__global__ void qproj_kernel(const float* __restrict__ cls,
                             const float* __restrict__ Wq,
                             const float* __restrict__ bq,
                             float* __restrict__ qbuf,
                             float* __restrict__ nqbuf) {
  int bh = blockIdx.x;
  int b = bh >> 3;
  int h = bh & 7;
  int n = h * 128 + threadIdx.x;
  const float* wrow = Wq + (size_t)n * Dc;
  const float* cl = cls + (size_t)b * Dc;
  float acc = bq[n];
  for (int k = 0; k < Dc; ++k) acc += cl[k] * wrow[k];
  qbuf[(size_t)b * Dc + n] = acc;

  float sq = acc * acc;
  for (int o = 16; o > 0; o >>= 1) sq += __shfl_xor(sq, o, 32);
  __shared__ float red[4];
  if ((threadIdx.x & 31) == 0) red[threadIdx.x >> 5] = sq;
  __syncthreads();
  if (threadIdx.x == 0)
    nqbuf[bh] = sqrtf(red[0] + red[1] + red[2] + red[3]);
}

// ---------------------------------------------------------------------------
// K/V projection GEMM via V_WMMA_F32_16X16X4_F32.
// Blocking: workgroup = 64 rows x 256 cols; wave = 64 rows x 32 cols
// (4 m-tiles x 2 n-tiles = 8 v8f accumulators).
// A tiles double-buffered in LDS (k-major, stride LSTR), staged either by
// async global->LDS b128 copies (CDNA5 path) or pipelined float4 loads;
// staging of tile kk+32 overlaps the 64 WMMAs of tile kk.
// ---------------------------------------------------------------------------
__global__ void proj_gemm_kernel(const float* __restrict__ x,
                                 const float* __restrict__ WtK,
                                 const float* __restrict__ WtV,
                                 const float* __restrict__ bK,
                                 const float* __restrict__ bV,
                                 float* __restrict__ Kbuf,
                                 float* __restrict__ Vbuf) {
  __shared__ __align__(16) float Xs[2][32 * LSTR];

  int rb = blockIdx.x;
  int b  = rb >> 5;            // T/64 = 32 row-blocks per batch
  int t0 = (rb & 31) << 6;
  int cb = blockIdx.y;         // 4 col-blocks of 256

  const float* Wt   = blockIdx.z ? WtV : WtK;
  const float* bias = blockIdx.z ? bV  : bK;
  float*       outp = blockIdx.z ? Vbuf : Kbuf;

  int tid  = threadIdx.x;
  int wave = tid >> 5;
  int lane = tid & 31;
  int m_l  = lane & 15;
  int hi   = lane >> 4;
  int nbase = cb * 256 + wave * 32;

  // Staging coordinates: thread copies two 4-float chunks (rows sk, sk+16).
  int sk = tid >> 4;           // k row 0..15
  int sm = (tid & 15) << 2;    // m 0..60 step 4
  const float* srcBase = x + (size_t)b * Dc * Tc + t0;

  v8f zero = {0.f, 0.f, 0.f, 0.f, 0.f, 0.f, 0.f, 0.f};
  v8f acc[4][2];
#pragma unroll
  for (int mi = 0; mi < 4; ++mi)
#pragma unroll
    for (int nt = 0; nt < 2; ++nt) acc[mi][nt] = zero;

#if USE_ASYNC_COPY
  // ---- async pipeline: prologue stage of tile kk=0 ----
  {
    const float* g0 = srcBase + (size_t)sk * Tc + sm;
    const float* g1 = srcBase + (size_t)(sk + 16) * Tc + sm;
    __builtin_amdgcn_global_load_async_to_lds_b128(
        ASYNC_GPTR(g0), ASYNC_SPTR(&Xs[0][sk * LSTR + sm]), 0, 0);
    __builtin_amdgcn_global_load_async_to_lds_b128(
        ASYNC_GPTR(g1), ASYNC_SPTR(&Xs[0][(sk + 16) * LSTR + sm]), 0, 0);
  }
  WAIT_ASYNC0();
  __syncthreads();
#else
  // ---- fallback pipeline: preload tile kk=0 into registers ----
  float4 p0 = *(const float4*)(srcBase + (size_t)sk * Tc + sm);
  float4 p1 = *(const float4*)(srcBase + (size_t)(sk + 16) * Tc + sm);
#endif

  for (int kk = 0; kk < Dc; kk += 32) {
    int cur = (kk >> 5) & 1;

#if USE_ASYNC_COPY
    if (kk + 32 < Dc) {   // issue next tile's copy; lands during compute
      const float* g0 = srcBase + (size_t)(kk + 32 + sk) * Tc + sm;
      const float* g1 = srcBase + (size_t)(kk + 32 + sk + 16) * Tc + sm;
      __builtin_amdgcn_global_load_async_to_lds_b128(
          ASYNC_GPTR(g0), ASYNC_SPTR(&Xs[cur ^ 1][sk * LSTR + sm]), 0, 0);
      __builtin_amdgcn_global_load_async_to_lds_b128(
          ASYNC_GPTR(g1), ASYNC_SPTR(&Xs[cur ^ 1][(sk + 16) * LSTR + sm]), 0, 0);
    }
#else
    *(float4*)&Xs[cur][sk * LSTR + sm]        = p0;
    *(float4*)&Xs[cur][(sk + 16) * LSTR + sm] = p1;
    __syncthreads();
    if (kk + 32 < Dc) {   // next tile's loads fly during compute
      p0 = *(const float4*)(srcBase + (size_t)(kk + 32 + sk) * Tc + sm);
      p1 = *(const float4*)(srcBase + (size_t)(kk + 32 + sk + 16) * Tc + sm);
    }
#endif

#pragma unroll
    for (int kb = 0; kb < 8; ++kb) {
      int kl = kb * 4 + hi * 2;           // even: selects a k-pair row
      v2f a[4];
#pragma unroll
      for (int mi = 0; mi < 4; ++mi) {
        a[mi].x = Xs[cur][kl * LSTR + mi * 16 + m_l];
        a[mi].y = Xs[cur][(kl + 1) * LSTR + mi * 16 + m_l];
      }
#pragma unroll
      for (int nt = 0; nt < 2; ++nt) {
        const v2f* Bp = (const v2f*)(Wt + ((size_t)(kk + kl) >> 1) * (2 * Dc)) +
                        (nbase + nt * 16 + m_l);
        v2f bf = *Bp;                     // single b64, coalesced over lanes
#pragma unroll
        for (int mi = 0; mi < 4; ++mi)
          acc[mi][nt] = WMMA_F32(a[mi], bf, acc[mi][nt]);
      }
    }

#if USE_ASYNC_COPY
    WAIT_ASYNC0();        // our async writes into the other buffer landed
    __syncthreads();      // everyone done reading cur / writing nxt
#endif
  }

  size_t rowbase = (size_t)b * Tc + t0;
#pragma unroll
  for (int mi = 0; mi < 4; ++mi) {
#pragma unroll
    for (int nt = 0; nt < 2; ++nt) {
      int n = nbase + nt * 16 + m_l;
      float bb = bias[n];
#pragma unroll
      for (int r = 0; r < 8; ++r) {
        int m = mi * 16 + r + hi * 8;
        outp[(rowbase + m) * Dc + n] = acc[mi][nt][r] + bb;
      }
    }
  }
}

// ---------------------------------------------------------------------------
// p_attn[b,h,t] = exp(|q.k / max(||q||*||k||, eps)|)
// ---------------------------------------------------------------------------
__global__ void attn_kernel(const float* __restrict__ Kbuf,
                            const float* __restrict__ qbuf,
                            const float* __restrict__ nq,
                            float* __restrict__ p_attn) {
  int bt = blockIdx.x;
  int b = bt >> 11;     // T = 2048
  int t = bt & 2047;
  int h = threadIdx.x >> 5;
  int lane = threadIdx.x & 31;

  const float* kp = Kbuf + (size_t)bt * Dc + h * 128 + lane * 4;
  const float* qp = qbuf + (size_t)b * Dc + h * 128 + lane * 4;
  float4 k4 = *(const float4*)kp;
  float4 q4 = *(const float4*)qp;
  float dot = k4.x * q4.x + k4.y * q4.y + k4.z * q4.z + k4.w * q4.w;
  float k2  = k4.x * k4.x + k4.y * k4.y + k4.z * k4.z + k4.w * k4.w;
  for (int o = 16; o > 0; o >>= 1) {
    dot += __shfl_xor(dot, o, 32);
    k2  += __shfl_xor(k2, o, 32);
  }
  if (lane == 0) {
    float denom = fmaxf(nq[b * Hc + h] * sqrtf(k2), 1e-8f);
    p_attn[(size_t)(b * Hc + h) * Tc + t] = expf(fabsf(dot / denom));
  }
}

// ---------------------------------------------------------------------------
// ROI adaptive average pooling with on-the-fly p_attn scaling.
// ---------------------------------------------------------------------------
__global__ void pool_kernel(const float* __restrict__ Vbuf,
                            const float* __restrict__ p_attn,
                            const int* __restrict__ rois,
                            float* __restrict__ P) {
  constexpr int nbs[4]  = {1, 3, 7, 9};
  constexpr long poff[4] = {0L,
                            (long)NROI * CG * 1,
                            (long)NROI * CG * 4,
                            (long)NROI * CG * 11};
  int nj = blockIdx.x;
  int n = nj >> 2;
  int j = nj & 3;
  int c = threadIdx.x;

  int bidx = rois[n * 3 + 0];
  int s    = rois[n * 3 + 1];
  int e    = rois[n * 3 + 2];
  int L    = e - s;
  int nb   = nbs[j];

  int h = j * 2 + (c >> 7);
  const float* pa = p_attn + (size_t)(bidx * Hc + h) * Tc;
  const float* vb = Vbuf + (size_t)bidx * Tc * Dc + j * CG + c;
  float* Pj = P + poff[j] + (size_t)n * (CG * nb) + (size_t)c * nb;

  for (int i = 0; i < nb; ++i) {
    int bs = s + (i * L) / nb;
    int be = s + ((i + 1) * L + nb - 1) / nb;   // ceil
    int cnt = be - bs;
    if (cnt < 1) cnt = 1;
    float acc = 0.f;
    for (int t = bs; t < be; ++t)
      acc += pa[t] * vb[(size_t)t * Dc];
    Pj[i] = acc / (float)cnt;
  }
}

// ---------------------------------------------------------------------------
// out[n, colOff + o] = sum_k A[n,k] * W[o,k] + bias[o]
// A row-major [128, K]; W in pair-interleaved layout [K/2][256][2].
// ---------------------------------------------------------------------------
__global__ void head_gemm_kernel(const float* __restrict__ A,
                                 const float* __restrict__ Wt,
                                 const float* __restrict__ bias,
                                 float* __restrict__ out,
                                 int K, int colOff) {
  __shared__ float Xs[16 * 33];

  int m0 = blockIdx.x * 16;
  int tid  = threadIdx.x;
  int wave = tid >> 5;
  int lane = tid & 31;
  int m_l  = lane & 15;
  int hi   = lane >> 4;

  v8f zero = {0.f, 0.f, 0.f, 0.f, 0.f, 0.f, 0.f, 0.f};
  v8f acc[2];
  acc[0] = zero;
  acc[1] = zero;

  int ks = tid & 31;
  int ms = tid >> 5;

  for (int kk = 0; kk < K; kk += 32) {
    float t0 = A[(size_t)(m0 + ms) * K + kk + ks];
    float t1 = A[(size_t)(m0 + ms + 8) * K + kk + ks];
    Xs[ms * 33 + ks]       = t0;
    Xs[(ms + 8) * 33 + ks] = t1;
    __syncthreads();

#pragma unroll
    for (int kb = 0; kb < 8; ++kb) {
      int kl = kb * 4 + hi * 2;
      v2f a;
      a.x = Xs[m_l * 33 + kl];
      a.y = Xs[m_l * 33 + kl + 1];
#pragma unroll
      for (int nt = 0; nt < 2; ++nt) {
        const v2f* Bp = (const v2f*)(Wt + ((size_t)(kk + kl) >> 1) * (2 * CG)) +
                        (wave * 32 + nt * 16 + m_l);
        v2f bf = *Bp;
        acc[nt] = WMMA_F32(a, bf, acc[nt]);
      }
    }
    __syncthreads();
  }

#pragma unroll
  for (int nt = 0; nt < 2; ++nt) {
    int c = wave * 32 + nt * 16 + m_l;
    float bb = bias[c];
#pragma unroll
    for (int r = 0; r < 8; ++r) {
      int m = r + hi * 8;
      out[(size_t)(m0 + m) * 1024 + colOff + c] = acc[nt][r] + bb;
    }
  }
}

// ---------------------------------------------------------------------------
// Host launch
// ---------------------------------------------------------------------------
extern "C" void kernel_launch(void* const* d_in, const int* in_sizes, int n_in,
                              void* d_out, int out_size, void* d_ws, size_t ws_size,
                              hipStream_t stream) {
  const float* x     = (const float*)d_in[0];   // [B, D, T]
  const float* cls   = (const float*)d_in[1];   // [B, D]
  const int*   rois  = (const int*)  d_in[2];   // [N, 3]
  const float* Wq    = (const float*)d_in[3];
  const float* bq    = (const float*)d_in[4];
  const float* Wk    = (const float*)d_in[5];
  const float* bk    = (const float*)d_in[6];
  const float* Wv    = (const float*)d_in[7];
  const float* bv    = (const float*)d_in[8];
  const float* convw[4] = {(const float*)d_in[9],  (const float*)d_in[11],
                           (const float*)d_in[13], (const float*)d_in[15]};
  const float* convb[4] = {(const float*)d_in[10], (const float*)d_in[12],
                           (const float*)d_in[14], (const float*)d_in[16]};
  float* out = (float*)d_out;                   // [128, 1024]

  // Workspace layout (floats)
  float* w = (float*)d_ws;
  const size_t DD = (size_t)Dc * Dc;            // 1M
  float* WtK  = w;                 // 1M (pair-interleaved)
  float* WtV  = WtK + DD;          // 1M (pair-interleaved)
  float* Kbuf = WtV + DD;          // B*T*D = 8M
  float* Vbuf = Kbuf + (size_t)Bc * Tc * Dc;    // 8M
  float* qbuf = Vbuf + (size_t)Bc * Tc * Dc;    // 4K
  float* nqb  = qbuf + (size_t)Bc * Dc;         // 32
  float* pa   = nqb + Bc * Hc;                  // B*H*T = 64K
  float* P    = pa + (size_t)Bc * Hc * Tc;      // 128*256*20
  float* Wtc  = P + (size_t)NROI * CG * 20;     // 256*(256+768+1792+2304)

  const int nbs[4] = {1, 3, 7, 9};
  const long PoffF[4] = {0L, (long)NROI * CG * 1, (long)NROI * CG * 4,
                         (long)NROI * CG * 11};
  long wtcOff[4];
  {
    long o = 0;
    for (int j = 0; j < 4; ++j) { wtcOff[j] = o; o += (long)CG * nbs[j] * CG; }
  }

  // 1) Weight transposes into k-pair-interleaved layout
  {
    int total = (int)DD;
    int blocks = (total + 255) / 256;
    transpose_pair_kernel<<<blocks, 256, 0, stream>>>(Wk, WtK, Dc, Dc);
    transpose_pair_kernel<<<blocks, 256, 0, stream>>>(Wv, WtV, Dc, Dc);
    for (int j = 0; j < 4; ++j) {
      int Kj = CG * nbs[j];
      int tj = CG * Kj;
      transpose_pair_kernel<<<(tj + 255) / 256, 256, 0, stream>>>(
          convw[j], Wtc + wtcOff[j], CG, Kj);
    }
  }

  // 2) q projection + norms
  qproj_kernel<<<Bc * Hc, 128, 0, stream>>>(cls, Wq, bq, qbuf, nqb);

  // 3) K and V projections (WMMA, async-staged double-buffered A tiles)
  proj_gemm_kernel<<<dim3((Bc * Tc) / 64, Dc / 256, 2), 256, 0, stream>>>(
      x, WtK, WtV, bk, bv, Kbuf, Vbuf);

  // 4) cosine attention weights
  attn_kernel<<<Bc * Tc, 256, 0, stream>>>(Kbuf, qbuf, nqb, pa);

  // 5) ROI adaptive average pooling (p_attn-scaled V)
  pool_kernel<<<NROI * 4, 256, 0, stream>>>(Vbuf, pa, rois, P);

  // 6) per-scale output contraction (WMMA)
  for (int j = 0; j < 4; ++j) {
    int Kj = CG * nbs[j];
    head_gemm_kernel<<<NROI / 16, 256, 0, stream>>>(
        P + PoffF[j], Wtc + wtcOff[j], convb[j], out, Kj, j * CG);
  }
}